// Loss_58695023067344
// MI455X (gfx1250) — compile-verified
//
#include <hip/hip_runtime.h>
#include <math.h>

#define TPB 256
#define NBOX 150

typedef float v2f __attribute__((ext_vector_type(2)));
typedef float v8f __attribute__((ext_vector_type(8)));
typedef int   v4i __attribute__((ext_vector_type(4)));

#define AS1 __attribute__((address_space(1)))
#define AS3 __attribute__((address_space(3)))

#if defined(__gfx1250__) && __has_builtin(__builtin_amdgcn_global_load_async_to_lds_b128)
#define HAVE_ASYNC_LDS 1
#else
#define HAVE_ASYNC_LDS 0
#endif

__device__ __forceinline__ void wait_asynccnt0() {
#if defined(__gfx1250__)
#if __has_builtin(__builtin_amdgcn_s_wait_asynccnt)
  __builtin_amdgcn_s_wait_asynccnt(0);
#else
  asm volatile("s_wait_asynccnt 0" ::: "memory");
#endif
#endif
}

#if HAVE_ASYNC_LDS
// 16B async global->LDS copy (GLOBAL_LOAD_ASYNC_TO_LDS_B128, ASYNCcnt path).
// Builtin takes typed v4i pointers: AS1 (printed as __device__) src, AS3 dst.
__device__ __forceinline__ void async_copy_b128(const float* gsrc, float* ldst) {
  __builtin_amdgcn_global_load_async_to_lds_b128(
      (AS1 v4i*)(v4i*)gsrc, (AS3 v4i*)(v4i*)ldst, 0, 0);
}
#endif

// Exact-f32 wave32 reduction using V_WMMA_F32_16X16X4_F32 with an all-ones
// A matrix: D[m][n] = sum_k B[k][n]. Each lane contributes its partial in
// B vgpr0 (b.x) and 0 in b.y, so D's 16 columns are the 16 pairwise sums of
// the 32 lane partials. d[0] lanes 0..15 hold those column sums (documented
// C/D layout); 4 shfl_down steps finish the sum into lane 0.
__device__ __forceinline__ float wave_reduce_wmma(float v) {
  v2f a; a.x = 1.0f; a.y = 1.0f;
  v2f b; b.x = v;    b.y = 0.0f;
  v8f c = {};
  v8f d = __builtin_amdgcn_wmma_f32_16x16x4_f32(false, a, false, b,
                                                (short)0, c, false, false);
  float s = d[0];
  s += __shfl_down(s, 8, 32);
  s += __shfl_down(s, 4, 32);
  s += __shfl_down(s, 2, 32);
  s += __shfl_down(s, 1, 32);
  return s; // valid in lane 0 of each wave
}

__global__ __launch_bounds__(TPB) void yolo_loss_cells(
    const float* __restrict__ out, const float* __restrict__ gt,
    const float* __restrict__ coords, int blocks_per_batch,
    float* __restrict__ partials, int partial_base) {
  __shared__ float s_tlx[NBOX], s_tly[NBOX], s_brx[NBOX], s_bry[NBOX],
      s_area[NBOX];
#if HAVE_ASYNC_LDS
  __shared__ float s_raw[NBOX * 4];
#endif
  __shared__ float s_wsum[TPB / 32];

  const int tid = threadIdx.x;
  const int batch = blockIdx.x / blocks_per_batch;
  const int blk_in_b = blockIdx.x - batch * blocks_per_batch;
  const int cell = batch * (blocks_per_batch * TPB) + blk_in_b * TPB + tid;

  const float* o = out + (size_t)cell * 5;
  const float* t = gt + (size_t)cell * 5;
  __builtin_prefetch(o, 0, 3);
  __builtin_prefetch(t, 0, 3);

  const float* cb = coords + (size_t)batch * (NBOX * 4);

  // ---- stage this batch's 150 gt boxes into LDS (corners + area, SoA) ----
#if HAVE_ASYNC_LDS
  if (tid < NBOX) {
    async_copy_b128(cb + tid * 4, &s_raw[tid * 4]);
  }
  wait_asynccnt0();  // this wave's async DMA into LDS is complete
  if (tid < NBOX) {
    float bx = s_raw[4 * tid + 0], by = s_raw[4 * tid + 1];
    float bw = s_raw[4 * tid + 2], bh = s_raw[4 * tid + 3];
    s_tlx[tid] = bx - 0.5f * bw;
    s_tly[tid] = by - 0.5f * bh;
    s_brx[tid] = bx + 0.5f * bw;
    s_bry[tid] = by + 0.5f * bh;
    s_area[tid] = bw * bh;
  }
#else
  if (tid < NBOX) {
    const float4 v = ((const float4*)cb)[tid];
    s_tlx[tid] = v.x - 0.5f * v.z;
    s_tly[tid] = v.y - 0.5f * v.w;
    s_brx[tid] = v.x + 0.5f * v.z;
    s_bry[tid] = v.y + 0.5f * v.w;
    s_area[tid] = v.z * v.w;
  }
#endif
  __syncthreads();

  // ---- per-cell loads ----
  const float ox = o[0], oy = o[1], ow = o[2], oh = o[3], oconf = o[4];
  const float tx = t[0], ty = t[1], tw = t[2], th = t[3], tconf = t[4];

  const float o_tlx = ox - 0.5f * ow, o_tly = oy - 0.5f * oh;
  const float o_brx = ox + 0.5f * ow, o_bry = oy + 0.5f * oh;
  const float g_tlx = tx - 0.5f * tw, g_tly = ty - 0.5f * th;
  const float g_brx = tx + 0.5f * tw, g_bry = ty + 0.5f * th;
  const float areaO = ow * oh, areaG = tw * th;

  // ---- DIoU(out, gt) ----
  float iw = fmaxf(fminf(o_brx, g_brx) - fmaxf(o_tlx, g_tlx), 0.0f);
  float ih = fmaxf(fminf(o_bry, g_bry) - fmaxf(o_tly, g_tly), 0.0f);
  float inter = iw * ih;
  float iou = inter / (areaO + areaG - inter + 1e-9f);
  float dx = ox - tx, dy = oy - ty;
  float cd2 = dx * dx + dy * dy;
  float ew = fmaxf(o_brx, g_brx) - fminf(o_tlx, g_tlx);
  float eh = fmaxf(o_bry, g_bry) - fminf(o_tly, g_tly);
  float diou = iou - cd2 / (ew * ew + eh * eh + 1e-9f);
  float box_scale = areaG * (1.0f / (512.0f * 512.0f));
  float giou_term = tconf * (2.0f - box_scale) * (1.0f - diou);

  // ---- background predicate: max_n IoU_n < 0.5, division-free.
  // inter/(union+eps) < 0.5  <=>  2*inter < union + eps  (union+eps > 0) ----
  bool all_below = true;
#pragma unroll 5
  for (int n = 0; n < NBOX; ++n) {
    float iw2 = fmaxf(fminf(o_brx, s_brx[n]) - fmaxf(o_tlx, s_tlx[n]), 0.0f);
    float ih2 = fmaxf(fminf(o_bry, s_bry[n]) - fmaxf(o_tly, s_tly[n]), 0.0f);
    float in2 = iw2 * ih2;
    float un2 = areaO + s_area[n] - in2;
    all_below = all_below && (2.0f * in2 < un2 + 1e-9f);
  }

  float background = (1.0f - tconf) * (all_below ? 1.0f : 0.0f);
  float p = fminf(fmaxf(oconf, 1e-7f), 1.0f - 1e-7f);
  float bce = -(tconf * logf(p) + (1.0f - tconf) * logf(1.0f - p));
  float dc = tconf - oconf;
  float focal = fabsf(tconf - 0.75f) * (dc * dc);
  float conf_term = focal * (tconf + background) * bce;

  float local = giou_term + conf_term;

  // ---- deterministic block reduction (WMMA f32 + wave32 shuffles) ----
  float w = wave_reduce_wmma(local);
  if ((tid & 31) == 0) s_wsum[tid >> 5] = w;
  __syncthreads();
  if (tid == 0) {
    float b = 0.0f;
#pragma unroll
    for (int i = 0; i < TPB / 32; ++i) b += s_wsum[i];
    partials[partial_base + blockIdx.x] = b;
  }
}

__global__ __launch_bounds__(TPB) void yolo_loss_finalize(
    const float* __restrict__ partials, int n, float* __restrict__ outp) {
  __shared__ float s_wsum[TPB / 32];
  float local = 0.0f;
  for (int i = threadIdx.x; i < n; i += TPB) local += partials[i];
  float w = wave_reduce_wmma(local);
  if ((threadIdx.x & 31) == 0) s_wsum[threadIdx.x >> 5] = w;
  __syncthreads();
  if (threadIdx.x == 0) {
    float s = 0.0f;
#pragma unroll
    for (int i = 0; i < TPB / 32; ++i) s += s_wsum[i];
    outp[0] = s * (1.0f / 16.0f);  // mean over B=16, shared by all 3 scales
  }
}

extern "C" void kernel_launch(void* const* d_in, const int* in_sizes, int n_in,
                              void* d_out, int out_size, void* d_ws,
                              size_t ws_size, hipStream_t stream) {
  const float* s_out = (const float*)d_in[0];
  const float* m_out = (const float*)d_in[1];
  const float* l_out = (const float*)d_in[2];
  const float* s_gt = (const float*)d_in[3];
  const float* m_gt = (const float*)d_in[4];
  const float* l_gt = (const float*)d_in[5];
  const float* s_coords = (const float*)d_in[6];
  const float* m_coords = (const float*)d_in[7];
  const float* l_coords = (const float*)d_in[8];

  float* partials = (float*)d_ws;

  // s: 16*64*64*3 = 196608 cells -> 768 blocks (48/batch)
  // m: 16*32*32*3 =  49152 cells -> 192 blocks (12/batch)
  // l: 16*16*16*3 =  12288 cells ->  48 blocks ( 3/batch)
  yolo_loss_cells<<<768, TPB, 0, stream>>>(s_out, s_gt, s_coords, 48, partials, 0);
  yolo_loss_cells<<<192, TPB, 0, stream>>>(m_out, m_gt, m_coords, 12, partials, 768);
  yolo_loss_cells<<<48, TPB, 0, stream>>>(l_out, l_gt, l_coords, 3, partials, 960);
  yolo_loss_finalize<<<1, TPB, 0, stream>>>(partials, 1008, (float*)d_out);
}